// GNNComm_29652454212127
// MI455X (gfx1250) — compile-verified
//
#include <hip/hip_runtime.h>

// Problem constants (match reference)
#define NMSG 2048
#define CD   8      // COMM_DIM
#define HID  64     // HIDDEN

typedef __attribute__((ext_vector_type(2))) float v2f;
typedef __attribute__((ext_vector_type(8))) float v8f;

// ---------------------------------------------------------------------------
// Kernel 1: projections via V_WMMA_F32_16X16X4_F32 (wave-per-tile, wave32).
//   A1[i,h] = sum_d msg[i,d] * W1[h, d]
//   A2[i,h] = sum_d msg[i,d] * W1[h, 8+d] + b1[h]
// W1 is [64 x 16] row-major. Two K=4 WMMA steps cover d=0..7, accumulator-
// chained in the v8f C operand. EXEC is all-ones (32 threads, no divergence).
// ---------------------------------------------------------------------------
__global__ __launch_bounds__(32) void gnn_proj_wmma(
    const float* __restrict__ msg, const float* __restrict__ W1,
    const float* __restrict__ b1,
    float* __restrict__ A1, float* __restrict__ A2) {
  const int lane = threadIdx.x;          // 0..31
  const int tile = blockIdx.x;           // 128 row-tiles x 4 col-tiles
  const int i0   = (tile >> 2) << 4;     // message-row tile base
  const int h0   = (tile & 3) << 4;      // hidden-col tile base
  const int half = lane >> 4;            // 0: K={0,1}; 1: K={2,3}
  const int l15  = lane & 15;

  v8f c1 = {}; // accumulates A1 tile
  v8f c2 = {}; // accumulates A2 tile

#pragma unroll
  for (int kc = 0; kc < 2; ++kc) {       // two K=4 chunks covering d=0..7
    const int k0 = kc * 4 + half * 2;
    // A 16x4 tile from messages: lane<16 holds K=k0,k0+1 of row M=l15
    v2f a;
    a.x = msg[(i0 + l15) * CD + k0 + 0];
    a.y = msg[(i0 + l15) * CD + k0 + 1];
    // B 4x16 tiles: B[k,n] = W1[h0+n, base + k]; same K striping as A
    v2f bL, bR;
    bL.x = W1[(h0 + l15) * (2 * CD) + k0 + 0];
    bL.y = W1[(h0 + l15) * (2 * CD) + k0 + 1];
    bR.x = W1[(h0 + l15) * (2 * CD) + CD + k0 + 0];
    bR.y = W1[(h0 + l15) * (2 * CD) + CD + k0 + 1];
    c1 = __builtin_amdgcn_wmma_f32_16x16x4_f32(false, a, false, bL,
                                               (short)0, c1, false, false);
    c2 = __builtin_amdgcn_wmma_f32_16x16x4_f32(false, a, false, bR,
                                               (short)0, c2, false, false);
  }

  // C/D layout: VGPR r -> M = r + 8*half, N = l15
#pragma unroll
  for (int r = 0; r < 8; ++r) {
    const int gi = i0 + r + half * 8;
    const int gh = h0 + l15;
    A1[gi * HID + gh] = c1[r];
    A2[gi * HID + gh] = c2[r] + b1[gh];
  }
}

// ---------------------------------------------------------------------------
// Kernel 2: fused pairwise relu-sum + diagonal correction + output GEMM.
// Block: 256 threads handle TI=16 i-rows x 64 h. Thread t: h = t&63,
// iq = t>>6 -> owns i = i0 + iq*4 + k, k=0..3 (4 accumulators per LDS read).
// j loop stages TJ=64 rows of A2 into LDS (16 KB of the 320 KB/WGP pool).
// ---------------------------------------------------------------------------
#define TI 16
#define TJ 64
__global__ __launch_bounds__(256) void gnn_pair_reduce(
    const float* __restrict__ A1, const float* __restrict__ A2,
    const float* __restrict__ W2, const float* __restrict__ b2,
    float* __restrict__ out) {
  __shared__ float sA2[TJ * HID];   // 16 KB staging for A2 chunk
  __shared__ float sSD[TI * HID];   // 4 KB for (S - D) epilogue

  const int tid = threadIdx.x;
  const int h   = tid & 63;
  const int iq  = tid >> 6;         // 0..3
  const int i0  = blockIdx.x * TI;

  float a[4], acc[4];
#pragma unroll
  for (int k = 0; k < 4; ++k) {
    a[k]   = A1[(i0 + iq * 4 + k) * HID + h];
    acc[k] = 0.0f;
  }

  for (int j0 = 0; j0 < NMSG; j0 += TJ) {
    __syncthreads();  // protect sA2 reuse
    // stage TJ*HID floats = 1024 float4; 4 per thread
    const float4* src = (const float4*)(A2 + j0 * HID);
    float4*       dst = (float4*)sA2;
#pragma unroll
    for (int v = 0; v < (TJ * HID / 4) / 256; ++v)
      dst[tid + v * 256] = src[tid + v * 256];
    __syncthreads();

    // prefetch next chunk toward L2/L0 while we burn VALU on this one
    if (j0 + TJ < NMSG)
      __builtin_prefetch(A2 + (j0 + TJ) * HID + tid * 16, 0, 0);

#pragma unroll 8
    for (int jj = 0; jj < TJ; ++jj) {
      const float c = sA2[jj * HID + h];   // conflict-free: h spans banks
#pragma unroll
      for (int k = 0; k < 4; ++k)
        acc[k] += fmaxf(a[k] + c, 0.0f);
    }
  }

  // subtract the j == i diagonal term, park S-D in LDS
#pragma unroll
  for (int k = 0; k < 4; ++k) {
    const float c = A2[(i0 + iq * 4 + k) * HID + h];
    acc[k] -= fmaxf(a[k] + c, 0.0f);
    sSD[(iq * 4 + k) * HID + h] = acc[k];
  }
  __syncthreads();

  // epilogue: out[i,d] = (S-D)[i,:] . W2[d,:] / (N-1) + b2[d]
  if (tid < TI * CD) {
    const int il = tid >> 3;
    const int d  = tid & 7;
    float s = 0.0f;
#pragma unroll
    for (int hh = 0; hh < HID; ++hh)
      s = fmaf(sSD[il * HID + hh], W2[d * HID + hh], s);
    out[(i0 + il) * CD + d] = s * (1.0f / (float)(NMSG - 1)) + b2[d];
  }
}

// ---------------------------------------------------------------------------
extern "C" void kernel_launch(void* const* d_in, const int* in_sizes, int n_in,
                              void* d_out, int out_size, void* d_ws, size_t ws_size,
                              hipStream_t stream) {
  const float* msg = (const float*)d_in[0];  // [2048, 8]
  const float* W1  = (const float*)d_in[1];  // [64, 16]
  const float* b1  = (const float*)d_in[2];  // [64]
  const float* W2  = (const float*)d_in[3];  // [8, 64]
  const float* b2  = (const float*)d_in[4];  // [8]
  float* out = (float*)d_out;                // [2048, 8]

  float* A1 = (float*)d_ws;                  // [2048, 64]
  float* A2 = A1 + NMSG * HID;               // [2048, 64] (total 1 MB of ws)

  // 128 row-tiles x 4 col-tiles, one wave32 per tile
  gnn_proj_wmma<<<dim3((NMSG / 16) * (HID / 16)), dim3(32), 0, stream>>>(
      msg, W1, b1, A1, A2);

  gnn_pair_reduce<<<dim3(NMSG / TI), dim3(256), 0, stream>>>(
      A1, A2, W2, b2, out);
}